// TensorProduct_2911987826944
// MI455X (gfx1250) — compile-verified
//
#include <hip/hip_runtime.h>
#include <hip/hip_bf16.h>

// out[z,k] = sum_{i,j} Mix[k,i,j] * f1[z,i] * f2[z,j]
// == GEMM: A(4096x3600) @ B(3600x3600), A[z,ij]=f1[z,i]*f2[z,j], B[ij,k]=Mix[k,ij]
// fp32 WMMA 16x16x4 + double-buffered LDS + CDNA5 async global->LDS copies.

typedef float  v2f  __attribute__((ext_vector_type(2)));
typedef float  v4f  __attribute__((ext_vector_type(4)));
typedef float  v8f  __attribute__((ext_vector_type(8)));

#define BM     128          // z-tile
#define BN     128          // k-out tile
#define BK     24           // ij chunk (3600 = 150 * 24)
#define BKP    28           // padded LDS row stride (112B: 16B-aligned, full bank spread)
#define NCHUNK 150
#define ZDIM   4096
#define NDIM   3600
#define KDIM   3600
#define FDIM   60

// CDNA5 async copy: global -> LDS, 16B per active lane, tracked by ASYNCcnt.
__device__ __forceinline__ void async_copy_b128(void* lds_ptr, const void* gptr) {
    unsigned lds_off = (unsigned)(size_t)lds_ptr;   // low 32 bits of flat addr == LDS offset
    asm volatile("global_load_async_to_lds_b128 %0, %1, off"
                 :: "v"(lds_off), "v"(gptr)
                 : "memory");
}
__device__ __forceinline__ void wait_async0() {
    asm volatile("s_wait_asynccnt 0" ::: "memory");
}

__global__ __launch_bounds__(256)
void tp_wmma_f32_kernel(const float* __restrict__ f1,
                        const float* __restrict__ f2,
                        const float* __restrict__ mix,   // (NDIM, KDIM) row-major
                        float* __restrict__ out)         // (ZDIM, NDIM)
{
    __shared__ float lds_A[2][BM * BKP];   // A tile: [z][ij]
    __shared__ float lds_B[2][BN * BKP];   // B^T tile: [n][ij] (mix rows contiguous in ij)

    const int tid  = threadIdx.x;
    const int lane = tid & 31;
    const int wave = tid >> 5;
    const int lh   = lane >> 4;     // lane half -> K pair within fragment
    const int l16  = lane & 15;     // M (A) / N (B,C) index within fragment

    const int nBase = blockIdx.x * BN;   // 29 tiles (29*128 >= 3600)
    const int zBase = blockIdx.y * BM;   // 32 tiles exactly

    const int wm = (wave >> 2) * 64;     // wave z-offset in block
    const int wn = (wave &  3) * 32;     // wave n-offset in block

    // ---- stage helpers ------------------------------------------------------
    // B^T tile rows rn=0..127, 24 floats each = 6 x b128; 768 b128 / 256 thr = 3 each
    auto stageB = [&](int kb, int buf) {
#pragma unroll
        for (int p = 0; p < 3; ++p) {
            int q  = tid + p * 256;     // 0..767
            int rn = q / 6;
            int c4 = q % 6;
            int n  = nBase + rn;
            if (n < NDIM)
                async_copy_b128(&lds_B[buf][rn * BKP + c4 * 4],
                                mix + (size_t)n * KDIM + kb + c4 * 4);
        }
    };
    // A tile: fused outer product, 3072 floats / 256 thr = 3 x float4 each
    auto stageA = [&](int kb, int buf) {
#pragma unroll
        for (int p = 0; p < 3; ++p) {
            int q  = tid + p * 256;
            int rz = q / 6;
            int c4 = q % 6;
            const float* r1 = f1 + (size_t)(zBase + rz) * FDIM;
            const float* r2 = f2 + (size_t)(zBase + rz) * FDIM;
            v4f v;
#pragma unroll
            for (int e = 0; e < 4; ++e) {
                int ij = kb + c4 * 4 + e;
                int i  = (int)((unsigned)ij / 60u);   // const-div -> mulhi
                int j  = ij - i * 60;
                v[e] = r1[i] * r2[j];
            }
            *(v4f*)(&lds_A[buf][rz * BKP + c4 * 4]) = v;
        }
    };

    v8f acc[4][2];
#pragma unroll
    for (int mt = 0; mt < 4; ++mt)
#pragma unroll
        for (int nt = 0; nt < 2; ++nt)
            acc[mt][nt] = (v8f){0.f,0.f,0.f,0.f,0.f,0.f,0.f,0.f};

    // ---- prologue: pre-zero OOB B rows (never async-loaded), stage chunk 0 --
    stageB(0, 0);
    stageA(0, 0);
    {
        int oobStart = NDIM - nBase;           // first OOB row in tile (may be >= BN)
        if (oobStart < BN) {
            int nOOB = BN - oobStart;
            for (int q = tid; q < nOOB * BKP * 2; q += 256) {
                int b = q / (nOOB * BKP);
                int r = (q / BKP) % nOOB;
                int c = q % BKP;
                lds_B[b][(oobStart + r) * BKP + c] = 0.f;
            }
        }
    }
    wait_async0();
    __syncthreads();

    // ---- main loop: async-stage chunk t+1 while WMMAs consume chunk t -------
    for (int t = 0; t < NCHUNK; ++t) {
        const int cur = t & 1;
        const int nxt = cur ^ 1;
        if (t + 1 < NCHUNK) {
            stageB((t + 1) * BK, nxt);
            stageA((t + 1) * BK, nxt);
        }

        const float* As = &lds_A[cur][0];
        const float* Bs = &lds_B[cur][0];
#pragma unroll
        for (int s = 0; s < BK / 4; ++s) {
            const int kk = s * 4 + lh * 2;   // per-lane K pair
            v2f afrag[4], bfrag[2];
#pragma unroll
            for (int mt = 0; mt < 4; ++mt)   // A 16x4: lane -> M=l16, vgprs -> K=kk,kk+1
                afrag[mt] = *(const v2f*)(As + (wm + mt * 16 + l16) * BKP + kk);
#pragma unroll
            for (int nt = 0; nt < 2; ++nt)   // B 4x16: lane -> N=l16, vgprs -> K=kk,kk+1
                bfrag[nt] = *(const v2f*)(Bs + (wn + nt * 16 + l16) * BKP + kk);
#pragma unroll
            for (int mt = 0; mt < 4; ++mt)
#pragma unroll
                for (int nt = 0; nt < 2; ++nt)
                    acc[mt][nt] = __builtin_amdgcn_wmma_f32_16x16x4_f32(
                        false, afrag[mt], false, bfrag[nt],
                        (short)0, acc[mt][nt], false, false);
        }

        wait_async0();       // chunk t+1 async copies landed in LDS
        __syncthreads();     // everyone done reading 'cur' / writing 'nxt'
    }

    // ---- store C: VGPR v of lane L holds (M = v + 8*(L/16), N = L%16) -------
#pragma unroll
    for (int mt = 0; mt < 4; ++mt) {
#pragma unroll
        for (int nt = 0; nt < 2; ++nt) {
            int n = nBase + wn + nt * 16 + l16;
            if (n < NDIM) {
#pragma unroll
                for (int v = 0; v < 8; ++v) {
                    int z = zBase + wm + mt * 16 + lh * 8 + v;
                    out[(size_t)z * NDIM + n] = acc[mt][nt][v];
                }
            }
        }
    }
}

extern "C" void kernel_launch(void* const* d_in, const int* in_sizes, int n_in,
                              void* d_out, int out_size, void* d_ws, size_t ws_size,
                              hipStream_t stream) {
    const float* f1  = (const float*)d_in[0];   // (4096, 60)
    const float* f2  = (const float*)d_in[1];   // (4096, 60)
    const float* mix = (const float*)d_in[2];   // (3600, 60, 60) == (3600, 3600)
    float* out = (float*)d_out;                 // (4096, 3600)
    (void)in_sizes; (void)n_in; (void)out_size; (void)d_ws; (void)ws_size;

    dim3 grid((NDIM + BN - 1) / BN, ZDIM / BM); // (29, 32)
    dim3 block(256);
    tp_wmma_f32_kernel<<<grid, block, 0, stream>>>(f1, f2, mix, out);
}